// GINLayer_15049565405785
// MI455X (gfx1250) — compile-verified
//
#include <hip/hip_runtime.h>
#include <hip/hip_bf16.h>

typedef float v2f __attribute__((ext_vector_type(2)));
typedef float v8f __attribute__((ext_vector_type(8)));

#define N_NODES 40000
#define H_DIM   128
#define N_EDGES 640000
#define LDS_STRIDE 132   // 132*4B = 528B = 33*16B -> rows stay 16B aligned, kills bank conflicts
#define BN_EPS 1e-5f

// ---------------------------------------------------------------------------
// 1) agg = (1 + eps) * x        (float4 vectorized, N*H/4 threads)
// ---------------------------------------------------------------------------
__global__ void gin_scale_init(const float* __restrict__ x,
                               const float* __restrict__ eps,
                               float* __restrict__ agg) {
    int i = blockIdx.x * blockDim.x + threadIdx.x;   // over N*H/4
    float s = 1.0f + eps[0];
    float4 v = ((const float4*)x)[i];
    v.x *= s; v.y *= s; v.z *= s; v.w *= s;
    ((float4*)agg)[i] = v;
}

// ---------------------------------------------------------------------------
// 2) scatter-add: one wave32 per edge, each lane carries a float4 of the row
// ---------------------------------------------------------------------------
__global__ void gin_scatter_add(const float* __restrict__ x,
                                const int* __restrict__ src,
                                const int* __restrict__ tgt,
                                float* __restrict__ agg) {
    int wave = (blockIdx.x * blockDim.x + threadIdx.x) >> 5;
    int lane = threadIdx.x & 31;
    if (wave >= N_EDGES) return;
    int s = src[wave];
    int t = tgt[wave];
    const float4 v = *(const float4*)(x + (size_t)s * H_DIM + lane * 4);
    float* dst = agg + (size_t)t * H_DIM + lane * 4;
    atomicAdd(dst + 0, v.x);
    atomicAdd(dst + 1, v.y);
    atomicAdd(dst + 2, v.z);
    atomicAdd(dst + 3, v.w);
}

// ---------------------------------------------------------------------------
// 3) zero the BN sum / sumsq accumulators (256 floats)
// ---------------------------------------------------------------------------
__global__ void gin_zero_stats(float* __restrict__ p) {
    p[threadIdx.x] = 0.0f;
}

// ---------------------------------------------------------------------------
// 4/5) GEMM: out[n,j] = act( A[n,:] . W[j,:] + bias[j] )
//      One block = 16-row panel (A staged in LDS), 8 waves = 8 col tiles,
//      32 x V_WMMA_F32_16X16X4_F32 over K=128.
//      BNSTATS: fused per-column sum / sumsq accumulation.
// ---------------------------------------------------------------------------
template<bool RELU, bool BNSTATS>
__global__ void gin_gemm_wmma(const float* __restrict__ A,
                              const float* __restrict__ W,
                              const float* __restrict__ bias,
                              float* __restrict__ out,
                              float* __restrict__ bn_sum,
                              float* __restrict__ bn_sumsq) {
    __shared__ float As[16 * LDS_STRIDE];

    const int mt   = blockIdx.x;        // row tile: 0..2499
    const int tid  = threadIdx.x;       // 0..255
    const int wave = tid >> 5;          // col tile: 0..7
    const int lane = tid & 31;

    // ---- cooperative, coalesced load of the 16x128 A panel into LDS ----
    {
        const float4* Ag = (const float4*)(A + (size_t)mt * 16 * H_DIM);
#pragma unroll
        for (int i = 0; i < 2; ++i) {
            int f    = tid + i * 256;   // 0..511 float4 slots (16 rows * 32)
            int row  = f >> 5;
            int colv = f & 31;
            float4 v = Ag[row * 32 + colv];
            *(float4*)(As + row * LDS_STRIDE + colv * 4) = v;
        }
    }
    __syncthreads();

    const int m  = lane & 15;           // A row in tile / B,D column in tile
    const int hi = lane >> 4;           // half-wave select: K pairs {0,1} vs {2,3}
    const float* Arow = As + m * LDS_STRIDE;
    const float* Wrow = W + (size_t)(wave * 16 + m) * H_DIM;  // B[k][j] = W[j][k]

    v8f c = {};
#pragma unroll
    for (int kt = 0; kt < 32; ++kt) {
        const int k = kt * 4 + hi * 2;
        v2f a; a.x = Arow[k]; a.y = Arow[k + 1];
        v2f b; b.x = Wrow[k]; b.y = Wrow[k + 1];
        // D = A(16x4,f32) x B(4x16,f32) + C   -> v_wmma_f32_16x16x4_f32
        c = __builtin_amdgcn_wmma_f32_16x16x4_f32(
                /*neg_a=*/false, a, /*neg_b=*/false, b,
                /*c_mod=*/(short)0, c, /*reuse_a=*/false, /*reuse_b=*/false);
    }

    // ---- epilogue: bias (+ReLU) store; optional fused BN partial stats ----
    const int col = wave * 16 + m;
    const float bv = bias[col];
    float s = 0.0f, sq = 0.0f;
#pragma unroll
    for (int r = 0; r < 8; ++r) {
        float v = c[r] + bv;
        if (RELU) v = fmaxf(v, 0.0f);
        out[(size_t)(mt * 16 + r + 8 * hi) * H_DIM + col] = v;
        if (BNSTATS) { s += v; sq += v * v; }
    }
    if (BNSTATS) {
        // lanes L and L+16 hold the same column (rows r and r+8): fold halves
        s  += __shfl_down(s, 16, 32);
        sq += __shfl_down(sq, 16, 32);
        if (hi == 0) {
            atomicAdd(&bn_sum[col], s);
            atomicAdd(&bn_sumsq[col], sq);
        }
    }
}

// ---------------------------------------------------------------------------
// 6) BatchNorm (train-mode, biased var) + ReLU
// ---------------------------------------------------------------------------
__global__ void gin_bn_relu(const float* __restrict__ h2,
                            const float* __restrict__ stats,  // [0:128)=sum, [128:256)=sumsq
                            const float* __restrict__ gamma,
                            const float* __restrict__ beta,
                            float* __restrict__ outp) {
    int i = blockIdx.x * blockDim.x + threadIdx.x;   // over N*H
    int col = i & (H_DIM - 1);
    const float invN = 1.0f / (float)N_NODES;
    float mean = stats[col] * invN;
    float ex2  = stats[H_DIM + col] * invN;
    float var  = ex2 - mean * mean;
    float inv  = __frsqrt_rn(var + BN_EPS);
    float v = (h2[i] - mean) * inv * gamma[col] + beta[col];
    outp[i] = fmaxf(v, 0.0f);
}

// ---------------------------------------------------------------------------
extern "C" void kernel_launch(void* const* d_in, const int* in_sizes, int n_in,
                              void* d_out, int out_size, void* d_ws, size_t ws_size,
                              hipStream_t stream) {
    const float* x     = (const float*)d_in[0];
    const int*   eidx  = (const int*)d_in[1];      // [2, E] flat: row0=src, row1=tgt
    const float* eps   = (const float*)d_in[2];
    const float* W1    = (const float*)d_in[3];
    const float* b1    = (const float*)d_in[4];
    const float* W2    = (const float*)d_in[5];
    const float* b2    = (const float*)d_in[6];
    const float* gamma = (const float*)d_in[7];
    const float* beta  = (const float*)d_in[8];
    float* outp = (float*)d_out;

    const size_t NH = (size_t)N_NODES * H_DIM;     // 5,120,000
    float* agg   = (float*)d_ws;                   // [N,H]
    float* h1    = agg + NH;                       // [N,H]
    float* h2    = h1 + NH;                        // [N,H]
    float* stats = h2 + NH;                        // 256 floats: sum | sumsq

    const int* src = eidx;
    const int* tgt = eidx + N_EDGES;

    // 1) agg = (1+eps)*x : NH/4 float4 elems, 256 thr -> 5000 blocks
    gin_scale_init<<<(int)(NH / 4 / 256), 256, 0, stream>>>(x, eps, agg);

    // 2) scatter-add: 640000 edges, 8 waves/block -> 80000 blocks? No: 640000/8
    gin_scatter_add<<<N_EDGES / 8, 256, 0, stream>>>(x, src, tgt, agg);

    // 3) zero BN stats
    gin_zero_stats<<<1, 256, 0, stream>>>(stats);

    // 4) h1 = relu(agg @ W1^T + b1) : 2500 row tiles
    gin_gemm_wmma<true, false><<<N_NODES / 16, 256, 0, stream>>>(
        agg, W1, b1, h1, nullptr, nullptr);

    // 5) h2 = h1 @ W2^T + b2, fused BN partial sums
    gin_gemm_wmma<false, true><<<N_NODES / 16, 256, 0, stream>>>(
        h1, W2, b2, h2, stats, stats + H_DIM);

    // 6) BN + ReLU -> d_out : NH elems, 256 thr -> 20000 blocks
    gin_bn_relu<<<(int)(NH / 256), 256, 0, stream>>>(h2, stats, gamma, beta, outp);
}